// SA_LUT3DGenerator_9612136808553
// MI455X (gfx1250) — compile-verified
//
#include <hip/hip_runtime.h>

// Problem constants (from reference): B=16, R=8, V=33, H=W=512, 3 colors.
#define B_      16
#define R_      8
#define V_      33
#define H_      512
#define W_      512
#define VCUBE   (V_ * V_ * V_)        // 35937
#define M_TOTAL (3 * VCUBE)           // 107811 rows of lut_bank_w
#define HW      (H_ * W_)             // 262144 pixels per image

typedef __attribute__((ext_vector_type(2))) float v2f;
typedef __attribute__((ext_vector_type(4))) float v4f;
typedef __attribute__((ext_vector_type(8))) float v8f;

// ---------------------------------------------------------------------------
// Kernel 1: LUT synthesis as a WMMA GEMM.
//   lut[b][m] = sum_r bank[m][r] * weights[b][r]
// Shapes: M = 107811 (tiled by 16), N = 16 (= B, single WMMA tile),
//         K = 8 (= R, two chained V_WMMA_F32_16X16X4_F32 accumulations).
// One wave per 16-row M tile; 8 waves (256 threads) per block.
// EXEC stays all-ones inside each executing wave (tile guard is wave-uniform),
// as required by WMMA.  Full tiles (all but the last) take an unpredicated
// path storing each lane's 8 contiguous floats as two global_store_b128.
// ---------------------------------------------------------------------------
__global__ void lut_gen_wmma(const float* __restrict__ weights, // (B_, R_)
                             const float* __restrict__ bank,    // (M_TOTAL, R_)
                             float* __restrict__ lut)           // (B_, M_TOTAL)
{
    const int lane = threadIdx.x & 31;
    const int wave = threadIdx.x >> 5;
    const int tile = blockIdx.x * (blockDim.x >> 5) + wave;
    const int m0   = tile * 16;
    if (m0 >= M_TOTAL) return;          // wave-uniform: EXEC all-1s for WMMA

    const int half = lane >> 4;         // 0: lanes 0-15, 1: lanes 16-31
    const int l16  = lane & 15;

    v8f acc = {};
    #pragma unroll
    for (int kb = 0; kb < R_; kb += 4) {
        // A (16x4, f32): lane half 0 holds K = kb+{0,1}, half 1 holds K = kb+{2,3},
        // M = l16 in both halves (ISA 7.12.2, 32-bit A 16x4 layout).
        int row    = m0 + l16;
        int rclamp = row < M_TOTAL ? row : (M_TOTAL - 1); // clamp; store is guarded
        const float* arow = bank + (long)rclamp * R_ + kb + half * 2;
        v2f a; a.x = arow[0]; a.y = arow[1];

        // B (4x16, f32): row-striped across lanes; VGPR j holds K = kb+j (half 0)
        // and K = kb+2+j (half 1), N = l16 = batch.  B[k][n] = weights[n*R_ + k].
        const float* bcol = weights + l16 * R_ + kb + half * 2;
        v2f bm; bm.x = bcol[0]; bm.y = bcol[1];

        // 8 args: (neg_a, A, neg_b, B, c_mod, C, reuse_a, reuse_b)
        acc = __builtin_amdgcn_wmma_f32_16x16x4_f32(
            false, a, false, bm, (short)0, acc, false, false);
    }

    // D (16x16 f32, C/D layout): VGPR j -> M = m0 + half*8 + j, N = l16 = batch.
    // Each lane owns 8 consecutive m values for one batch.
    const int   batch = l16;
    const int   mbase = m0 + half * 8;
    float*      dst   = lut + (long)batch * M_TOTAL + mbase;

    if (m0 + 16 <= M_TOTAL) {
        // Full tile (6738 of 6739): two unpredicated 16-byte vector stores.
        v4f lo = { acc[0], acc[1], acc[2], acc[3] };
        v4f hi = { acc[4], acc[5], acc[6], acc[7] };
        *(v4f*)(dst)     = lo;
        *(v4f*)(dst + 4) = hi;
    } else {
        // Edge tile only (3 valid rows): guarded scalar stores.
        #pragma unroll
        for (int j = 0; j < 8; ++j) {
            if (mbase + j < M_TOTAL) dst[j] = acc[j];
        }
    }
}

// ---------------------------------------------------------------------------
// Kernel 2: trilinear 3D-LUT apply (memory-bound: ~100 MB stream @ 23.3 TB/s
// -> ~4.5 us floor).  float4 I/O per thread; 24 gathers/pixel hit the 431 KB
// per-batch LUT slice which is L2/WGP$ resident.
// ---------------------------------------------------------------------------
__device__ __forceinline__ int imin_(int a, int b) { return a < b ? a : b; }

__device__ __forceinline__ void tri1(const float* __restrict__ L,
                                     float r, float g, float bl,
                                     float* __restrict__ o)
{
    const float s = (float)(V_ - 1);
    float cx = fminf(fmaxf(r  * s, 0.0f), s);
    float cy = fminf(fmaxf(g  * s, 0.0f), s);
    float cz = fminf(fmaxf(bl * s, 0.0f), s);
    int x0 = imin_((int)cx, V_ - 1);     // cx >= 0 so trunc == floor
    int y0 = imin_((int)cy, V_ - 1);
    int z0 = imin_((int)cz, V_ - 1);
    int x1 = imin_(x0 + 1, V_ - 1);
    int y1 = imin_(y0 + 1, V_ - 1);
    int z1 = imin_(z0 + 1, V_ - 1);
    float fx = cx - (float)x0, fy = cy - (float)y0, fz = cz - (float)z0;
    float gx = 1.0f - fx, gy = 1.0f - fy, gz = 1.0f - fz;

    // idx = (z*V + y)*V + x  (matches reference gather)
    int zy00 = (z0 * V_ + y0) * V_;
    int zy01 = (z0 * V_ + y1) * V_;
    int zy10 = (z1 * V_ + y0) * V_;
    int zy11 = (z1 * V_ + y1) * V_;
    int i000 = zy00 + x0, i001 = zy00 + x1, i010 = zy01 + x0, i011 = zy01 + x1;
    int i100 = zy10 + x0, i101 = zy10 + x1, i110 = zy11 + x0, i111 = zy11 + x1;

    float w000 = gz * gy * gx, w001 = gz * gy * fx;
    float w010 = gz * fy * gx, w011 = gz * fy * fx;
    float w100 = fz * gy * gx, w101 = fz * gy * fx;
    float w110 = fz * fy * gx, w111 = fz * fy * fx;

    #pragma unroll
    for (int c = 0; c < 3; ++c) {
        const float* Lc = L + c * VCUBE;
        o[c] = Lc[i000] * w000 + Lc[i001] * w001 + Lc[i010] * w010 + Lc[i011] * w011
             + Lc[i100] * w100 + Lc[i101] * w101 + Lc[i110] * w110 + Lc[i111] * w111;
    }
}

__global__ void lut_apply(const float* __restrict__ img,  // (B_, 3, H_, W_)
                          const float* __restrict__ lut,  // (B_, M_TOTAL)
                          float* __restrict__ out)        // (B_, 3, H_, W_)
{
    const int  b  = blockIdx.y;
    const int  q  = blockIdx.x * blockDim.x + threadIdx.x;  // quad index (exact fit)
    const long ib = (long)b * 3 * HW;
    const float* __restrict__ L = lut + (long)b * M_TOTAL;

    float4 r4 = ((const float4*)(img + ib         ))[q];
    float4 g4 = ((const float4*)(img + ib +     HW))[q];
    float4 b4 = ((const float4*)(img + ib + 2 * HW))[q];

    float4 o0, o1, o2;
    float  o[3];
    tri1(L, r4.x, g4.x, b4.x, o); o0.x = o[0]; o1.x = o[1]; o2.x = o[2];
    tri1(L, r4.y, g4.y, b4.y, o); o0.y = o[0]; o1.y = o[1]; o2.y = o[2];
    tri1(L, r4.z, g4.z, b4.z, o); o0.z = o[0]; o1.z = o[1]; o2.z = o[2];
    tri1(L, r4.w, g4.w, b4.w, o); o0.w = o[0]; o1.w = o[1]; o2.w = o[2];

    ((float4*)(out + ib         ))[q] = o0;
    ((float4*)(out + ib +     HW))[q] = o1;
    ((float4*)(out + ib + 2 * HW))[q] = o2;
}

// ---------------------------------------------------------------------------
// Launch: inputs in setup_inputs() order: weights, img, lut_bank_w.
// d_ws holds the synthesized per-batch LUTs: 16 * 107811 * 4 B ~= 6.6 MB.
// ---------------------------------------------------------------------------
extern "C" void kernel_launch(void* const* d_in, const int* in_sizes, int n_in,
                              void* d_out, int out_size, void* d_ws, size_t ws_size,
                              hipStream_t stream) {
    (void)in_sizes; (void)n_in; (void)out_size; (void)ws_size;
    const float* weights = (const float*)d_in[0];   // (B,R,1,1) -> (16,8)
    const float* img     = (const float*)d_in[1];   // (16,3,512,512)
    const float* bank    = (const float*)d_in[2];   // (107811, 8)
    float*       out     = (float*)d_out;           // (16,3,512,512)
    float*       lut     = (float*)d_ws;            // (16, 107811) scratch

    // Stage 1: WMMA GEMM -> per-batch LUTs. 6739 M-tiles, 8 waves/block.
    const int tiles = (M_TOTAL + 15) / 16;          // 6739
    dim3 blk1(256);
    dim3 grd1((tiles + 7) / 8);
    lut_gen_wmma<<<grd1, blk1, 0, stream>>>(weights, bank, lut);

    // Stage 2: trilinear apply. HW/4 = 65536 quads per batch, 256 thr/block.
    dim3 blk2(256);
    dim3 grd2(HW / 4 / 256, B_);                    // (256, 16)
    lut_apply<<<grd2, blk2, 0, stream>>>(img, lut, out);
}